// CRF_60112362275454
// MI455X (gfx1250) — compile-verified
//
#include <hip/hip_runtime.h>
#include <math.h>

// Problem constants (match reference)
#define S_LEN  256
#define B_SZ   64
#define D_DIM  1024
#define T_TAGS 64
#define BOS_T  0
#define EOS_T  1
#define PAD_T  2

typedef float v2f __attribute__((ext_vector_type(2)));
typedef float v8f __attribute__((ext_vector_type(8)));

// ---------------------------------------------------------------------------
// Kernel 1: emit[r,t] = sum_d features[r,d] * W[t,d] + b[t]
// r = s*B + b (row-major flatten of (S,B)), so this is a 16384 x 1024 x 64
// FP32 GEMM. One wave (wave32) computes a 16-row x 64-col strip using
// V_WMMA_F32_16X16X4_F32, reusing the A fragment across 4 N-tiles.
//
// A frag (16x4, MxK): lanes 0-15 hold {K=k, k+1} for M=lane;
//                     lanes 16-31 hold {K=k+2, k+3} for M=lane-16.
// B frag (4x16, KxN): lanes 0-15 hold {K=k, k+1} for N=lane;
//                     lanes 16-31 hold {K=k+2, k+3} for N=lane-16.
// C/D (16x16 f32): VGPR v, lanes 0-15 -> (M=v, N=lane);
//                  lanes 16-31 -> (M=v+8, N=lane-16).
// ---------------------------------------------------------------------------
__global__ __launch_bounds__(256) void crf_emit_gemm_wmma(
    const float* __restrict__ features,   // (S*B, D)
    const float* __restrict__ W,          // (T, D)
    const float* __restrict__ bias,       // (T)
    float* __restrict__ emit)             // (S*B, T)
{
    const int lane  = threadIdx.x & 31;
    const int wave  = threadIdx.x >> 5;
    const int tileM = blockIdx.x * 8 + wave;     // 16-row strip index
    const int m15   = lane & 15;
    const int khalf = lane >> 4;                 // 0 -> K=k..k+1, 1 -> K=k+2..k+3

    const float* aptr = features + (size_t)(tileM * 16 + m15) * D_DIM + khalf * 2;
    const float* b0   = W + (size_t)( 0 + m15) * D_DIM + khalf * 2;
    const float* b1   = W + (size_t)(16 + m15) * D_DIM + khalf * 2;
    const float* b2   = W + (size_t)(32 + m15) * D_DIM + khalf * 2;
    const float* b3   = W + (size_t)(48 + m15) * D_DIM + khalf * 2;

    v8f acc0 = {}, acc1 = {}, acc2 = {}, acc3 = {};

    for (int k = 0; k < D_DIM; k += 4) {
        v2f a  = *(const v2f*)(aptr + k);
        v2f f0 = *(const v2f*)(b0 + k);
        v2f f1 = *(const v2f*)(b1 + k);
        v2f f2 = *(const v2f*)(b2 + k);
        v2f f3 = *(const v2f*)(b3 + k);
        acc0 = __builtin_amdgcn_wmma_f32_16x16x4_f32(false, a, false, f0, (short)0, acc0, false, false);
        acc1 = __builtin_amdgcn_wmma_f32_16x16x4_f32(false, a, false, f1, (short)0, acc1, false, false);
        acc2 = __builtin_amdgcn_wmma_f32_16x16x4_f32(false, a, false, f2, (short)0, acc2, false, false);
        acc3 = __builtin_amdgcn_wmma_f32_16x16x4_f32(false, a, false, f3, (short)0, acc3, false, false);
    }

    const int rowC = tileM * 16 + khalf * 8;
    const int colC = m15;
    #pragma unroll
    for (int v = 0; v < 8; ++v) {
        size_t r = (size_t)(rowC + v) * T_TAGS;
        emit[r +  0 + colC] = acc0[v] + bias[ 0 + colC];
        emit[r + 16 + colC] = acc1[v] + bias[16 + colC];
        emit[r + 32 + colC] = acc2[v] + bias[32 + colC];
        emit[r + 48 + colC] = acc3[v] + bias[48 + colC];
    }
}

// ---------------------------------------------------------------------------
// Kernel 2: gold path score per batch element.
// gold[b] = sum_{s=0}^{len-1} emit[s,b,tags[s,b]] + trans[tags[s,b], nxt]
//           with nxt = tags[s+1] (or PAD at s == S-1).
// ---------------------------------------------------------------------------
__global__ __launch_bounds__(64) void crf_gold(
    const float* __restrict__ emit,
    const int*   __restrict__ tags,
    const int*   __restrict__ seq_lens,
    const float* __restrict__ trans,
    float* __restrict__ gold_part)
{
    const int b = threadIdx.x;
    const int len = seq_lens[b];
    float g = 0.0f;
    for (int s = 0; s < len; ++s) {
        int t0 = tags[s * B_SZ + b];
        int t1 = (s + 1 < S_LEN) ? tags[(s + 1) * B_SZ + b] : PAD_T;
        g += emit[((size_t)s * B_SZ + b) * T_TAGS + t0] + trans[t0 * T_TAGS + t1];
    }
    gold_part[b] = g;
}

// ---------------------------------------------------------------------------
// Kernel 3: forward algorithm. One block per batch element, one thread per
// destination tag j. Transitions transposed into LDS with stride 65 to avoid
// bank conflicts. Sequential over s (true dependency); active steps only
// (inactive steps leave cur unchanged, so we stop at len).
//   cur0[j]   = emit[0,b,BOS] + trans[BOS,j]
//   nxt[j]    = logsumexp_i( cur[i] + emit[s,b,i] + trans[i,j] )
// ---------------------------------------------------------------------------
__global__ __launch_bounds__(64) void crf_forward(
    const float* __restrict__ emit,
    const int*   __restrict__ seq_lens,
    const float* __restrict__ trans,
    float* __restrict__ alpha_part)
{
    __shared__ float tT[T_TAGS * (T_TAGS + 1)];   // tT[j*65 + i] = trans[i][j]
    __shared__ float u[T_TAGS];                   // cur[i] + emit[s,b,i]

    const int b = blockIdx.x;
    const int j = threadIdx.x;

    // Cooperative transposed load of transitions (coalesced global reads).
    for (int idx = j; idx < T_TAGS * T_TAGS; idx += T_TAGS) {
        int i  = idx >> 6;
        int jj = idx & 63;
        tT[jj * (T_TAGS + 1) + i] = trans[idx];
    }

    const int len = seq_lens[b];
    float curj = emit[(size_t)b * T_TAGS + BOS_T] + trans[BOS_T * T_TAGS + j];

    const float* tr = &tT[j * (T_TAGS + 1)];
    for (int s = 1; s < len; ++s) {
        u[j] = curj + emit[((size_t)s * B_SZ + b) * T_TAGS + j];
        if (s + 1 < S_LEN)
            __builtin_prefetch(&emit[((size_t)(s + 1) * B_SZ + b) * T_TAGS + j], 0, 0);
        __syncthreads();

        float m = -INFINITY;
        #pragma unroll 8
        for (int i = 0; i < T_TAGS; ++i) m = fmaxf(m, u[i] + tr[i]);
        float sum = 0.0f;
        #pragma unroll 8
        for (int i = 0; i < T_TAGS; ++i) sum += __expf(u[i] + tr[i] - m);
        float nxt = m + __logf(sum);

        __syncthreads();   // all reads of u done before next iteration rewrites it
        curj = nxt;
    }

    if (j == EOS_T) alpha_part[b] = curj;
}

// ---------------------------------------------------------------------------
// Kernel 4: deterministic tree reduction -> scalar result.
// out = sum_b alpha_part[b] - sum_b gold_part[b]
// ---------------------------------------------------------------------------
__global__ __launch_bounds__(64) void crf_finalize(
    const float* __restrict__ alpha_part,
    const float* __restrict__ gold_part,
    float* __restrict__ out)
{
    __shared__ float red[B_SZ];
    const int t = threadIdx.x;
    red[t] = alpha_part[t] - gold_part[t];
    __syncthreads();
    for (int off = 32; off > 0; off >>= 1) {
        if (t < off) red[t] += red[t + off];
        __syncthreads();
    }
    if (t == 0) out[0] = red[0];
}

extern "C" void kernel_launch(void* const* d_in, const int* in_sizes, int n_in,
                              void* d_out, int out_size, void* d_ws, size_t ws_size,
                              hipStream_t stream) {
    (void)in_sizes; (void)n_in; (void)out_size; (void)ws_size;

    const float* features = (const float*)d_in[0];   // (S,B,D) f32
    const int*   tags     = (const int*)  d_in[1];   // (S,B)   i32
    const int*   seq_lens = (const int*)  d_in[2];   // (B,)    i32
    const float* W        = (const float*)d_in[3];   // (T,D)   f32
    const float* bias     = (const float*)d_in[4];   // (T,)    f32
    const float* trans    = (const float*)d_in[5];   // (T,T)   f32
    float* out = (float*)d_out;

    float* emit       = (float*)d_ws;                          // S*B*T floats (4 MB)
    float* gold_part  = emit + (size_t)S_LEN * B_SZ * T_TAGS;  // B floats
    float* alpha_part = gold_part + B_SZ;                      // B floats

    // 16384 rows / (16 rows per wave * 8 waves per block) = 128 blocks
    crf_emit_gemm_wmma<<<dim3(128), dim3(256), 0, stream>>>(features, W, bias, emit);
    crf_gold<<<dim3(1), dim3(B_SZ), 0, stream>>>(emit, tags, seq_lens, trans, gold_part);
    crf_forward<<<dim3(B_SZ), dim3(T_TAGS), 0, stream>>>(emit, seq_lens, trans, alpha_part);
    crf_finalize<<<dim3(1), dim3(B_SZ), 0, stream>>>(alpha_part, gold_part, out);
}